// MHCMapping_38757784879745
// MI455X (gfx1250) — compile-verified
//
#include <hip/hip_runtime.h>
#include <hip/hip_bf16.h>
#include <math.h>

typedef __attribute__((ext_vector_type(2))) float v2f;
typedef __attribute__((ext_vector_type(4))) float v4f;
typedef __attribute__((ext_vector_type(8))) float v8f;
typedef __attribute__((ext_vector_type(4))) int   v4i;

#define TOKENS            16384     // B*T = 4*4096
#define K_DIM             8192      // n*C = 4*2048
#define N_OUT             24        // 4 (pre) + 4 (post) + 16 (res)
#define BK                64        // K-block staged in LDS
#define WAVES_PER_BLOCK   4
#define TOKENS_PER_WAVE   16
#define TOKENS_PER_BLOCK  (WAVES_PER_BLOCK * TOKENS_PER_WAVE)  // 64
#define LDS_STRIDE        68        // 64 + 4 pad floats -> conflict-free ds_load_b64
#define SK_ITERS          20

// ---- CDNA5 async global->LDS path (guarded; falls back to sync copy) ----
#if defined(__gfx1250__) && __has_builtin(__builtin_amdgcn_global_load_async_to_lds_b128)
#define USE_ASYNC_LDS 1
#else
#define USE_ASYNC_LDS 0
#endif

#if USE_ASYNC_LDS
typedef __attribute__((address_space(1))) v4i* g4p;   // global int4*
typedef __attribute__((address_space(3))) v4i* l4p;   // LDS int4*
#endif

struct Tiles {
    float Wt[32][LDS_STRIDE];                    // weight K-block (rows 24..31 stay zero)
    float At[WAVES_PER_BLOCK][16][LDS_STRIDE];   // per-wave x K-block
};

struct SmemT {
    Tiles buf[2];                                // double buffer
    float stage[TOKENS_PER_BLOCK][25];           // epilogue staging (24 + pad)
};

// 16-byte global -> LDS copy. cpol: 0 = RT (weights, L2-resident), 1 = NT (x stream, read-once).
template <int CPOL>
__device__ __forceinline__ void copy16(const float* g, float* l) {
#if USE_ASYNC_LDS
    __builtin_amdgcn_global_load_async_to_lds_b128(
        (g4p)(v4i*)g, (l4p)(v4i*)l, 0, CPOL);
#else
    *(v4f*)l = *(const v4f*)g;
#endif
}

__device__ __forceinline__ void wait_async0() {
#if USE_ASYNC_LDS
#if __has_builtin(__builtin_amdgcn_s_wait_asynccnt)
    __builtin_amdgcn_s_wait_asynccnt(0);
#else
    asm volatile("s_wait_asynccnt 0x0" ::: "memory");
#endif
#endif
}

__device__ __forceinline__ void load_tiles(Tiles* b,
                                           const float* __restrict__ x,
                                           const float* __restrict__ W_pre,
                                           const float* __restrict__ W_post,
                                           const float* __restrict__ W_res,
                                           int kb, int tid, int wave, int lane, int blockTok) {
    // Weight K-block: rows 0..23 x 64 floats = 384 b128 segments over 128 threads (3 iters).
    // 16 consecutive lanes cover one 256B row -> coalesced.
    #pragma unroll
    for (int i = 0; i < 3; ++i) {
        int seg  = tid + i * 128;          // 0..383
        int row  = seg >> 4;               // 0..23
        int colb = (seg & 15) << 2;        // 0,4,...,60
        const float* src;
        if (row < 4)      src = W_pre  + (size_t)row       * K_DIM + kb + colb;
        else if (row < 8) src = W_post + (size_t)(row - 4) * K_DIM + kb + colb;
        else              src = W_res  + (size_t)(row - 8) * K_DIM + kb + colb;
        copy16<0>(src, &b->Wt[row][colb]);
    }
    // Per-wave x K-block: 16 tokens x 64 floats = 256 segments over 32 lanes (8 iters).
    const float* xw = x + (size_t)(blockTok + wave * TOKENS_PER_WAVE) * K_DIM + kb;
    #pragma unroll
    for (int i = 0; i < 8; ++i) {
        int seg  = lane + i * 32;          // 0..255
        int row  = seg >> 4;               // 0..15
        int colb = (seg & 15) << 2;
        copy16<1>(xw + (size_t)row * K_DIM + colb, &b->At[wave][row][colb]);
    }
}

__device__ __forceinline__ float fast_sigmoid(float x) {
    return 1.0f / (1.0f + __expf(-x));
}

__global__ __launch_bounds__(128) void mhc_kernel(
    const float* __restrict__ x,
    const float* __restrict__ a_pre, const float* __restrict__ a_post, const float* __restrict__ a_res,
    const float* __restrict__ b_pre, const float* __restrict__ b_post, const float* __restrict__ b_res,
    const float* __restrict__ W_pre, const float* __restrict__ W_post, const float* __restrict__ W_res,
    float* __restrict__ out)
{
    __shared__ SmemT s;

    const int tid       = threadIdx.x;
    const int wave      = tid >> 5;
    const int lane      = tid & 31;
    const int laneMod16 = lane & 15;
    const int laneHi    = lane >> 4;                 // 0 or 1
    const int blockTok  = blockIdx.x * TOKENS_PER_BLOCK;

    // Zero the padded weight rows (24..31) once; load loop never touches them.
    #pragma unroll
    for (int bsel = 0; bsel < 2; ++bsel)
        for (int i = tid; i < 8 * LDS_STRIDE; i += 128)
            s.buf[bsel].Wt[24 + i / LDS_STRIDE][i % LDS_STRIDE] = 0.0f;

    v8f acc0 = {};   // outputs 0..15
    v8f acc1 = {};   // outputs 16..31 (16..23 valid)

    // Preload first K-block into buffer 0.
    load_tiles(&s.buf[0], x, W_pre, W_post, W_res, 0, tid, wave, lane, blockTok);

    int cur = 0;
    for (int kb = 0; kb < K_DIM; kb += BK) {
        wait_async0();        // this wave's async copies into buf[cur] have landed in LDS
        __syncthreads();      // all waves' copies landed; all waves done reading buf[cur^1]

        if (kb + BK < K_DIM)  // prefetch next K-block into the other buffer
            load_tiles(&s.buf[cur ^ 1], x, W_pre, W_post, W_res,
                       kb + BK, tid, wave, lane, blockTok);

        // ---- matrix core over buf[cur]: K=4 per v_wmma_f32_16x16x4_f32 ----
        // A 16x4: lane%16 = M row; VGPR{0,1} = K{0,1} (lanes 0-15) / K{2,3} (lanes 16-31).
        // B 4x16: lane%16 = N col; same K striping.
        Tiles* tb = &s.buf[cur];
        #pragma unroll
        for (int kk = 0; kk < BK; kk += 4) {
            const int kc = kk + 2 * laneHi;
            v2f a  = *(const v2f*)&tb->At[wave][laneMod16][kc];
            v2f b0 = *(const v2f*)&tb->Wt[laneMod16][kc];
            v2f b1 = *(const v2f*)&tb->Wt[16 + laneMod16][kc];
            acc0 = __builtin_amdgcn_wmma_f32_16x16x4_f32(
                false, a, false, b0, (short)0, acc0, false, false);
            acc1 = __builtin_amdgcn_wmma_f32_16x16x4_f32(
                false, a, false, b1, (short)0, acc1, false, false);
        }
        cur ^= 1;
    }

    __syncthreads();

    // ---- stage C tiles to LDS: lane holds N = lane%16 (+16 for acc1);
    //      VGPR r -> token (wave*16 + r + 8*laneHi) ----
    #pragma unroll
    for (int r = 0; r < 8; ++r) {
        int lt = wave * 16 + r + 8 * laneHi;
        s.stage[lt][laneMod16] = acc0[r];
        int n1 = 16 + laneMod16;
        if (n1 < N_OUT) s.stage[lt][n1] = acc1[r];
    }
    __syncthreads();

    // ---- epilogue: one thread per token ----
    if (tid < TOKENS_PER_BLOCK) {
        const int tglob = blockTok + tid;
        float y[N_OUT];
        #pragma unroll
        for (int i = 0; i < N_OUT; ++i) y[i] = s.stage[tid][i];

        const float sa_pre  = a_pre[0];
        const float sa_post = a_post[0];
        const float sa_res  = a_res[0];

        float* out_pre  = out;                       // [16384,4]
        float* out_post = out + TOKENS * 4;          // [16384,4]
        float* out_res  = out + TOKENS * 8;          // [16384,16]

        #pragma unroll
        for (int i = 0; i < 4; ++i) {
            float hp = sa_pre  * y[i]     + b_pre[i];
            float hq = sa_post * y[4 + i] + b_post[i];
            out_pre [tglob * 4 + i] = fast_sigmoid(hp);
            out_post[tglob * 4 + i] = 2.0f * fast_sigmoid(hq);
        }

        // log-domain Sinkhorn-Knopp on the 4x4 res logits
        float la[16];
        #pragma unroll
        for (int j = 0; j < 16; ++j) la[j] = sa_res * y[8 + j] + b_res[j];

        for (int it = 0; it < SK_ITERS; ++it) {
            #pragma unroll
            for (int i = 0; i < 4; ++i) {            // row normalize
                float* r = &la[4 * i];
                float m = fmaxf(fmaxf(r[0], r[1]), fmaxf(r[2], r[3]));
                float ssum = __expf(r[0] - m) + __expf(r[1] - m) +
                             __expf(r[2] - m) + __expf(r[3] - m);
                float lse = m + __logf(ssum);
                r[0] -= lse; r[1] -= lse; r[2] -= lse; r[3] -= lse;
            }
            #pragma unroll
            for (int j = 0; j < 4; ++j) {            // col normalize
                float m = fmaxf(fmaxf(la[j], la[4 + j]), fmaxf(la[8 + j], la[12 + j]));
                float ssum = __expf(la[j]       - m) + __expf(la[4 + j]  - m) +
                             __expf(la[8 + j]  - m) + __expf(la[12 + j] - m);
                float lse = m + __logf(ssum);
                la[j] -= lse; la[4 + j] -= lse; la[8 + j] -= lse; la[12 + j] -= lse;
            }
        }
        #pragma unroll
        for (int j = 0; j < 16; ++j) out_res[tglob * 16 + j] = __expf(la[j]);
    }
}

extern "C" void kernel_launch(void* const* d_in, const int* in_sizes, int n_in,
                              void* d_out, int out_size, void* d_ws, size_t ws_size,
                              hipStream_t stream) {
    (void)in_sizes; (void)n_in; (void)d_ws; (void)ws_size; (void)out_size;
    const float* x      = (const float*)d_in[0];
    const float* a_pre  = (const float*)d_in[1];
    const float* a_post = (const float*)d_in[2];
    const float* a_res  = (const float*)d_in[3];
    const float* b_pre  = (const float*)d_in[4];
    const float* b_post = (const float*)d_in[5];
    const float* b_res  = (const float*)d_in[6];
    const float* W_pre  = (const float*)d_in[7];
    const float* W_post = (const float*)d_in[8];
    const float* W_res  = (const float*)d_in[9];
    float* out = (float*)d_out;

    dim3 grid(TOKENS / TOKENS_PER_BLOCK);   // 256 workgroups
    dim3 block(128);                        // 4 waves (wave32)
    mhc_kernel<<<grid, block, 0, stream>>>(x, a_pre, a_post, a_res,
                                           b_pre, b_post, b_res,
                                           W_pre, W_post, W_res, out);
}